// RoiResizeConv_33809982554791
// MI455X (gfx1250) — compile-verified
//
#include <hip/hip_runtime.h>
#include <hip/hip_bf16.h>

// RoiResize (bilinear ROI pooling), MI455X / gfx1250.
// img: (1,128,128,1024) f32 NHWC, rois: (1,256,4) f32, out: (1,256,14,14,1024) f32.
// Memory-bound (AI ~1.7 flop/byte; img is 64 MB -> L2-resident, HBM ~270 MB => ~12 us floor).
// CDNA5 paths: double-buffered GLOBAL_LOAD_ASYNC_TO_LDS_B128 pipeline with
// depth-based s_wait_asynccnt (in-order async completion), b128 everywhere, wave32.

static constexpr int kH = 128;
static constexpr int kW = 128;
static constexpr int kC = 1024;
static constexpr int kR = 256;
static constexpr int kP = 14;

__global__ __launch_bounds__(256) void roi_resize_cdna5(
    const float* __restrict__ img,
    const float* __restrict__ rois,
    float* __restrict__ out)
{
    // Double-buffered staging: 2 x (4 neighbor rows x 1024 ch) x f32 = 32 KB.
    __shared__ float s_v[2][4][kC];

    const int blk = blockIdx.x;        // r * 14 + py
    const int py  = blk % kP;
    const int r   = blk / kP;
    const int t   = threadIdx.x;       // 0..255, one float4 of channels

    // ROI box (integers stored as f32; truncation == reference astype(int32)).
    const float4 rb = reinterpret_cast<const float4*>(rois)[r];
    const int x1 = (int)rb.x, y1 = (int)rb.y, x2 = (int)rb.z, y2 = (int)rb.w;
    const int hI = y2 - y1;
    const int wI = x2 - x1;

    // One divide each per block (reference: i * (h/14), i * (w/14)).
    const float qy = (float)hI / (float)kP;
    const float qx = (float)wI / (float)kP;

    // y-metadata: uniform, computed once per block.
    const float src_y = (float)py * qy;
    const int   y0    = (int)floorf(src_y);
    const float fy    = src_y - (float)y0;
    const int   y1i   = min(y0 + 1, hI - 1);
    const int   ay0   = min(max(y1 + y0, 0), kH - 1);
    const int   ay1   = min(max(y1 + y1i, 0), kH - 1);

    const unsigned cb     = (unsigned)t * 16u;                        // this lane's float4 (bytes)
    const unsigned rowB0  = (unsigned)(ay0 * kW) * (unsigned)kC * 4u; // row ay0 base (bytes)
    const unsigned rowB1  = (unsigned)(ay1 * kW) * (unsigned)kC * 4u; // row ay1 base (bytes)
    const unsigned long long gbase = (unsigned long long)(const void*)img;

    // x-metadata for one px (uniform per block).
    auto xmeta = [&](int px, int& ax0, int& ax1, float& fx) {
        const float src_x = (float)px * qx;
        const int   x0    = (int)floorf(src_x);
        fx = src_x - (float)x0;
        const int x1i = min(x0 + 1, wI - 1);
        ax0 = min(max(x1 + x0, 0), kW - 1);
        ax1 = min(max(x1 + x1i, 0), kW - 1);
    };

    // Issue the 4 async global->LDS b128 copies for cell (r, py, px) into buffer `buf`.
    // GVS addressing: saddr = image base (SGPR pair), vaddr = 32-bit per-lane byte offset.
    auto issue = [&](int px, int buf) {
        int ax0, ax1; float fx_unused;
        xmeta(px, ax0, ax1, fx_unused);
        const unsigned o00 = rowB0 + (unsigned)ax0 * (unsigned)(kC * 4) + cb;
        const unsigned o01 = rowB0 + (unsigned)ax1 * (unsigned)(kC * 4) + cb;
        const unsigned o10 = rowB1 + (unsigned)ax0 * (unsigned)(kC * 4) + cb;
        const unsigned o11 = rowB1 + (unsigned)ax1 * (unsigned)(kC * 4) + cb;
        // Flat shared pointers keep the LDS offset in addr[31:0] (CDNA5 aperture rule).
        const unsigned l00 = (unsigned)(unsigned long long)(const void*)&s_v[buf][0][4 * t];
        const unsigned l01 = (unsigned)(unsigned long long)(const void*)&s_v[buf][1][4 * t];
        const unsigned l10 = (unsigned)(unsigned long long)(const void*)&s_v[buf][2][4 * t];
        const unsigned l11 = (unsigned)(unsigned long long)(const void*)&s_v[buf][3][4 * t];
        asm volatile("global_load_async_to_lds_b128 %0, %1, %2"
                     :: "v"(l00), "v"(o00), "s"(gbase) : "memory");
        asm volatile("global_load_async_to_lds_b128 %0, %1, %2"
                     :: "v"(l01), "v"(o01), "s"(gbase) : "memory");
        asm volatile("global_load_async_to_lds_b128 %0, %1, %2"
                     :: "v"(l10), "v"(o10), "s"(gbase) : "memory");
        asm volatile("global_load_async_to_lds_b128 %0, %1, %2"
                     :: "v"(l11), "v"(o11), "s"(gbase) : "memory");
    };

    // Prologue: stage px = 0.
    issue(0, 0);

    for (int px = 0; px < kP; ++px) {
        const int cur = px & 1;

        if (px + 1 < kP) {
            // Stage the next cell, then wait until only its 4 copies remain in
            // flight: async loads complete in issue order, so asynccnt <= 4
            // guarantees the current cell's group has landed in LDS.
            issue(px + 1, cur ^ 1);
            asm volatile("s_wait_asynccnt 0x4" ::: "memory");
        } else {
            asm volatile("s_wait_asynccnt 0x0" ::: "memory");
        }

        int ax0_u, ax1_u; float fx;
        xmeta(px, ax0_u, ax1_u, fx);

        // Each lane consumes exactly the bytes it staged: no barrier needed.
        const float4 v00 = *reinterpret_cast<const float4*>(&s_v[cur][0][4 * t]);
        const float4 v01 = *reinterpret_cast<const float4*>(&s_v[cur][1][4 * t]);
        const float4 v10 = *reinterpret_cast<const float4*>(&s_v[cur][2][4 * t]);
        const float4 v11 = *reinterpret_cast<const float4*>(&s_v[cur][3][4 * t]);

        float4 o;
        {
            float tx, bx;
            tx = v00.x + (v01.x - v00.x) * fx;  bx = v10.x + (v11.x - v10.x) * fx;  o.x = tx + (bx - tx) * fy;
            tx = v00.y + (v01.y - v00.y) * fx;  bx = v10.y + (v11.y - v10.y) * fx;  o.y = tx + (bx - tx) * fy;
            tx = v00.z + (v01.z - v00.z) * fx;  bx = v10.z + (v11.z - v10.z) * fx;  o.z = tx + (bx - tx) * fy;
            tx = v00.w + (v01.w - v00.w) * fx;  bx = v10.w + (v11.w - v10.w) * fx;  o.w = tx + (bx - tx) * fy;
        }

        // Coalesced b128 store: this cell's 4 KB channel row is contiguous.
        const size_t cell = (size_t)blk * kP + px;   // ((r*14)+py)*14 + px
        reinterpret_cast<float4*>(out)[cell * (kC / 4) + t] = o;
    }
}

extern "C" void kernel_launch(void* const* d_in, const int* in_sizes, int n_in,
                              void* d_out, int out_size, void* d_ws, size_t ws_size,
                              hipStream_t stream) {
    const float* img  = (const float*)d_in[0];   // (1,128,128,1024) f32
    const float* rois = (const float*)d_in[1];   // (1,256,4) f32
    float* out = (float*)d_out;                  // (1,256,14,14,1024) f32

    const int blocks = kR * kP;                  // 3,584 blocks: one (roi, py) row each
    roi_resize_cdna5<<<dim3(blocks), dim3(256), 0, stream>>>(img, rois, out);
}